// Het_GNN_38603166056757
// MI455X (gfx1250) — compile-verified
//
#include <hip/hip_runtime.h>

#define NB   256   // partial-sum blocks per branch
#define DIN  200   // embed dim
#define OUT1 300
#define OUT2 400

typedef float v2f __attribute__((ext_vector_type(2)));
typedef float v8f __attribute__((ext_vector_type(8)));

// -------- Stage A: deterministic column partial sums over H[nrows, 200] ------
// grid = (NB, 2): blockIdx.y selects user/post branch.
// Thread t<250: fixed float4 column-group c4=t%50, sweeps 4 rows/iter.
__global__ __launch_bounds__(256) void colsum_partial(
    const float* __restrict__ user_H, const float* __restrict__ post_H,
    float* __restrict__ part, int nrows)
{
    const int t = threadIdx.x;
    const float4* __restrict__ H4 =
        (const float4*)(blockIdx.y ? post_H : user_H);
    __shared__ float lds[250 * 4];

    if (t < 250) {
        const int c4   = t % 50;      // float4 column group (covers cols 4*c4..4*c4+3)
        const int rsub = t / 50;      // 0..4
        float4 acc = make_float4(0.f, 0.f, 0.f, 0.f);
        const int stride = NB * 20;   // rows consumed per grid sweep
        for (int r0 = blockIdx.x * 20 + rsub * 4; r0 < nrows; r0 += stride) {
#pragma unroll
            for (int i = 0; i < 4; ++i) {
                const int r = r0 + i;
                if (r < nrows) {
                    float4 v = H4[r * 50 + c4];
                    acc.x += v.x; acc.y += v.y; acc.z += v.z; acc.w += v.w;
                }
            }
        }
        lds[t * 4 + 0] = acc.x; lds[t * 4 + 1] = acc.y;
        lds[t * 4 + 2] = acc.z; lds[t * 4 + 3] = acc.w;
    }
    __syncthreads();
    if (t < DIN) {                    // fixed-order combine of the 5 row-subsets
        const int c4 = t >> 2, comp = t & 3;
        float s = 0.f;
#pragma unroll
        for (int rs = 0; rs < 5; ++rs) s += lds[(rs * 50 + c4) * 4 + comp];
        part[(blockIdx.y * NB + blockIdx.x) * DIN + t] = s;
    }
}

// -------- Stage B: mean finalize + WMMA MLP head + softmax + sigmoid ---------
// Single block, 512 threads = 16 waves (wave32).
__global__ __launch_bounds__(512) void mlp_head(
    const float* __restrict__ part,
    const float* __restrict__ W1u, const float* __restrict__ b1u,
    const float* __restrict__ W2u, const float* __restrict__ b2u,
    const float* __restrict__ W1p, const float* __restrict__ b1p,
    const float* __restrict__ W2p, const float* __restrict__ b2p,
    const float* __restrict__ Wout, const float* __restrict__ bout,
    float* __restrict__ out, float invN)
{
    __shared__ float m[2][DIN];
    __shared__ float g1[2][OUT1];
    __shared__ float g2[2][OUT2];
    __shared__ float red[512];

    const int t    = threadIdx.x;
    const int lane = t & 31;
    const int wave = t >> 5;          // 0..15

    // Phase 1: finalize column means (fixed-order sum of NB partials).
    if (t < 2 * DIN) {
        const int br = t / DIN, c = t % DIN;
        float s = 0.f;
        for (int b = 0; b < NB; ++b) s += part[(br * NB + b) * DIN + c];
        m[br][c] = s * invN;
    }
    __syncthreads();

    // Common branchless lane roles for V_WMMA_F32_16X16X4_F32:
    //   A (16x4 f32): lanes 0-15 = rows M, VGPR0/1 = K0,K1 (lanes<16) or K2,K3 (>=16).
    //   Only row M=0 is populated (vector-matrix product) -> lanes 0 and 16 carry data.
    const int   kh    = (lane < 16) ? 0 : 2;
    const float amask = ((lane & 15) == 0) ? 1.f : 0.f;

    // Phase 2: g1 = relu(m @ W1 + b1), per-wave 16-col tiles.
    {
        const int ntile = (OUT1 + 15) / 16;       // 19
        for (int task = wave; task < 2 * ntile; task += 16) {
            const int br   = task / ntile;
            const int tile = task % ntile;
            const float* __restrict__ W1 = br ? W1p : W1u;
            const int col   = tile * 16 + (lane & 15);
            const int valid = col < OUT1;
            const int colc  = valid ? col : (OUT1 - 1);   // clamped, load unconditional
            v8f c8 = {};
#pragma unroll
            for (int k0 = 0; k0 < DIN; k0 += 4) {
                // all lanes load (8B-aligned LDS), mask by lane role — no EXEC branches
                const float2 mv = *(const float2*)&m[br][k0 + kh];
                v2f a; a.x = amask * mv.x; a.y = amask * mv.y;
                const float bx = W1[(k0 + kh)     * OUT1 + colc];
                const float by = W1[(k0 + kh + 1) * OUT1 + colc];
                v2f b; b.x = valid ? bx : 0.f; b.y = valid ? by : 0.f;
                c8 = __builtin_amdgcn_wmma_f32_16x16x4_f32(
                         false, a, false, b, (short)0, c8, false, false);
            }
            if (lane < 16 && valid) {             // D row 0 = VGPR0, lanes 0-15
                const float* b1 = br ? b1p : b1u;
                float v = c8[0] + b1[col];
                g1[br][col] = v > 0.f ? v : 0.f;
            }
        }
    }
    __syncthreads();

    // Phase 3: g2 = g1 @ W2 + b2  (K=300, 25 exact 16-col tiles per branch).
    {
        const int ntile = OUT2 / 16;              // 25
        for (int task = wave; task < 2 * ntile; task += 16) {
            const int br   = task / ntile;
            const int tile = task % ntile;
            const float* __restrict__ W2 = br ? W2p : W2u;
            const int col = tile * 16 + (lane & 15);
            v8f c8 = {};
#pragma unroll
            for (int k0 = 0; k0 < OUT1; k0 += 4) {
                const float2 gv = *(const float2*)&g1[br][k0 + kh];
                v2f a; a.x = amask * gv.x; a.y = amask * gv.y;
                v2f b;
                b.x = W2[(k0 + kh)     * OUT2 + col];
                b.y = W2[(k0 + kh + 1) * OUT2 + col];
                c8 = __builtin_amdgcn_wmma_f32_16x16x4_f32(
                         false, a, false, b, (short)0, c8, false, false);
            }
            if (lane < 16) {
                const float* b2 = br ? b2p : b2u;
                g2[br][col] = c8[0] + b2[col];
            }
        }
    }
    __syncthreads();

    // Phase 4: per-branch softmax over 400, combine, dot W_out, sigmoid.
    const int half = t >> 8;          // branch handled by this half of the block
    const int lt   = t & 255;

    float v = -3.4e38f;
    for (int j = lt; j < OUT2; j += 256) v = fmaxf(v, g2[half][j]);
    red[t] = v;
    __syncthreads();
    for (int s = 128; s > 0; s >>= 1) {
        if (lt < s) red[t] = fmaxf(red[t], red[t + s]);
        __syncthreads();
    }
    const float mx = red[half << 8];
    __syncthreads();

    float sum = 0.f;
    for (int j = lt; j < OUT2; j += 256) {
        float e = expf(g2[half][j] - mx);
        g2[half][j] = e;              // each (half,j) owned by exactly one thread
        sum += e;
    }
    red[t] = sum;
    __syncthreads();
    for (int s = 128; s > 0; s >>= 1) {
        if (lt < s) red[t] += red[t + s];
        __syncthreads();
    }
    const float s0 = red[0];
    const float s1 = red[256];
    __syncthreads();

    float dot = 0.f;
    if (t < OUT2) dot = (g2[0][t] / s0 + g2[1][t] / s1) * Wout[t];
    red[t] = dot;
    __syncthreads();
    for (int s = 256; s > 0; s >>= 1) {
        if (t < s) red[t] += red[t + s];
        __syncthreads();
    }
    if (t == 0) {
        const float z = red[0] + bout[0];
        out[0] = 1.f / (1.f + expf(-z));
    }
}

extern "C" void kernel_launch(void* const* d_in, const int* in_sizes, int n_in,
                              void* d_out, int out_size, void* d_ws, size_t ws_size,
                              hipStream_t stream) {
    const float* user_H = (const float*)d_in[0];
    const float* post_H = (const float*)d_in[1];
    const float* W1u = (const float*)d_in[2];
    const float* b1u = (const float*)d_in[3];
    const float* W2u = (const float*)d_in[4];
    const float* b2u = (const float*)d_in[5];
    const float* W1p = (const float*)d_in[6];
    const float* b1p = (const float*)d_in[7];
    const float* W2p = (const float*)d_in[8];
    const float* b2p = (const float*)d_in[9];
    const float* Wout = (const float*)d_in[10];
    const float* bout = (const float*)d_in[11];

    float* part = (float*)d_ws;                 // 2*NB*200 floats = 400 KB
    const int nrows = in_sizes[0] / DIN;        // 200000

    dim3 gA(NB, 2);
    colsum_partial<<<gA, 256, 0, stream>>>(user_H, post_H, part, nrows);
    mlp_head<<<1, 512, 0, stream>>>(part,
                                    W1u, b1u, W2u, b2u,
                                    W1p, b1p, W2p, b2p,
                                    Wout, bout,
                                    (float*)d_out, 1.0f / (float)nrows);
}